// RNN2Layer_30846455120440
// MI455X (gfx1250) — compile-verified
//
#include <hip/hip_runtime.h>
#include <cstddef>

#define T_STEPS 1024
#define BATCH   64
#define IN_DIM  256
#define OUT_DIM 256
#define H_DIM   1024
#define EUL     0.01f
#define NWG     16      // one workgroup per 64-wide H slice
#define LDSLD   68      // padded LDS leading dim (64 + 4) to dodge bank conflicts

typedef float v2f __attribute__((ext_vector_type(2)));
typedef float v8f __attribute__((ext_vector_type(8)));

// fp32 tensor op: D(16x16) = A(16x4) * B(4x16) + C   -> v_wmma_f32_16x16x4_f32
__device__ __forceinline__ v8f wmma_f32_k4(v2f a, v2f b, v8f c) {
  return __builtin_amdgcn_wmma_f32_16x16x4_f32(false, a, false, b, (short)0, c,
                                               false, false);
}

// ---------------------------------------------------------------------------
// Generic fp32 GEMM: C[M,N] = A[M,K] @ B[K,N].  64x64 tile / workgroup,
// 256 threads = 8 wave32, each wave owns two 16x16 WMMA accumulators
// (same M rows -> A fragments shared).  A staged row-major, B staged
// TRANSPOSED (N-major) so both A and B fragments are contiguous 64-bit
// LDS loads (no v_mov packing between WMMAs).
// ---------------------------------------------------------------------------
__global__ __launch_bounds__(256) void gemm64_f32(const float* __restrict__ A,
                                                  const float* __restrict__ B,
                                                  float* __restrict__ C,
                                                  int M, int N, int K,
                                                  int lda, int ldb, int ldc) {
  __shared__ float As[64 * LDSLD];   // As[m][k]
  __shared__ float Bs[64 * LDSLD];   // Bs[n][k]  (transposed)
  const int tid   = threadIdx.x;
  const int lane  = tid & 31;
  const int wave  = tid >> 5;
  const int tileM = blockIdx.x * 64;
  const int tileN = blockIdx.y * 64;

  const int m15 = lane & 15;            // A: M row / B,C: N col within 16
  const int kb  = (lane >> 4) << 1;     // lanes 16-31 hold K=2,3 of each K4
  const int mhi = (lane >> 4) * 8;      // C/D: lanes 16-31 hold M = r+8
  const int tr0 = ((wave * 2)     >> 2) * 16, tc0 = ((wave * 2)     & 3) * 16;
  const int tr1 = ((wave * 2 + 1) >> 2) * 16, tc1 = ((wave * 2 + 1) & 3) * 16;

  v8f acc0 = {0.f, 0.f, 0.f, 0.f, 0.f, 0.f, 0.f, 0.f};
  v8f acc1 = acc0;

  for (int k0 = 0; k0 < K; k0 += 64) {
    for (int i = tid; i < 64 * 64; i += 256) {
      const int r = i >> 6, c = i & 63;                  // c fastest: coalesced
      As[r * LDSLD + c] = A[(size_t)(tileM + r) * lda + (k0 + c)];
      Bs[c * LDSLD + r] = B[(size_t)(k0 + r) * ldb + (tileN + c)];
    }
    if (k0 + 64 < K) {  // warm caches for the next K slab -> global_prefetch_b8
      __builtin_prefetch(&A[(size_t)(tileM + (tid >> 2)) * lda + k0 + 64], 0, 1);
      __builtin_prefetch(&B[(size_t)(k0 + 64 + (tid >> 2)) * ldb + tileN], 0, 1);
    }
    __syncthreads();
#pragma unroll 4
    for (int kk = 0; kk < 64; kk += 4) {
      v2f a0, b0, b1;
      a0.x = As[(tr0 + m15) * LDSLD + kk + kb];
      a0.y = As[(tr0 + m15) * LDSLD + kk + kb + 1];
      b0.x = Bs[(tc0 + m15) * LDSLD + kk + kb];
      b0.y = Bs[(tc0 + m15) * LDSLD + kk + kb + 1];
      acc0 = wmma_f32_k4(a0, b0, acc0);
      b1.x = Bs[(tc1 + m15) * LDSLD + kk + kb];
      b1.y = Bs[(tc1 + m15) * LDSLD + kk + kb + 1];
      acc1 = wmma_f32_k4(a0, b1, acc1);   // tr0 == tr1: A fragment reused
    }
    __syncthreads();
  }

#pragma unroll
  for (int r = 0; r < 8; ++r) {
    C[(size_t)(tileM + tr0 + r + mhi) * ldc + tileN + tc0 + m15] = acc0[r];
    C[(size_t)(tileM + tr1 + r + mhi) * ldc + tileN + tc1 + m15] = acc1[r];
  }
}

// ---------------------------------------------------------------------------
// Persistent Euler-scan kernel.  16 workgroups; WG j owns H columns
// [64j, 64j+64).  emb[] enters holding proj = input @ W_in and is rewritten
// in place with the post-step states (== embeddings output).
// Per step:  acc = h_{t-1} @ C[:,slice]  (K=1024, fp32 WMMA)
//          + relu(h_{t-1}[:,slice]) @ B_blk[j]  (K=64, block-diagonal)
//   h_t[:,slice] = (1-eps)*h_{t-1}[:,slice] + eps*(acc + proj_t[:,slice])
// followed by a device-scope barrier (per-step counters in d_ws).
// ---------------------------------------------------------------------------
__global__ __launch_bounds__(256) void rnn_scan(const float* __restrict__ Bm,
                                                const float* __restrict__ Cm,
                                                float* __restrict__ emb,
                                                unsigned* __restrict__ cnt) {
  __shared__ float Sa[64 * LDSLD];   // Sa[m][k]  (row-major A source)
  __shared__ float Cb[64 * LDSLD];   // Cb[n][k]  (transposed B source)
  const int tid  = threadIdx.x;
  const int lane = tid & 31;
  const int wave = tid >> 5;
  const int jc   = blockIdx.x * 64;

  const int m15 = lane & 15;
  const int kb  = (lane >> 4) << 1;
  const int mhi = (lane >> 4) * 8;
  const int tr[2] = {((wave * 2) >> 2) * 16, ((wave * 2 + 1) >> 2) * 16};
  const int tc[2] = {((wave * 2) & 3) * 16, ((wave * 2 + 1) & 3) * 16};

  const v8f zero8 = {0.f, 0.f, 0.f, 0.f, 0.f, 0.f, 0.f, 0.f};

  for (int t = 0; t < T_STEPS; ++t) {
    v8f acc[2] = {zero8, zero8};

    if (t > 0) {
      const float* Sprev = emb + (size_t)(t - 1) * BATCH * H_DIM;
      // ---- coupling term: full K=1024 over h_{t-1}
      for (int k0 = 0; k0 < H_DIM; k0 += 64) {
        for (int i = tid; i < 64 * 64; i += 256) {
          const int r = i >> 6, c = i & 63;
          Sa[r * LDSLD + c] = Sprev[(size_t)r * H_DIM + k0 + c];
          Cb[c * LDSLD + r] = Cm[(size_t)(k0 + r) * H_DIM + jc + c];
        }
        if (k0 + 64 < H_DIM) {
          __builtin_prefetch(&Cm[(size_t)(k0 + 64 + (tid >> 2)) * H_DIM + jc],
                             0, 1);
        }
        __syncthreads();
        for (int kk = 0; kk < 64; kk += 4) {
          v2f a, b0, b1;
          a.x  = Sa[(tr[0] + m15) * LDSLD + kk + kb];
          a.y  = Sa[(tr[0] + m15) * LDSLD + kk + kb + 1];
          b0.x = Cb[(tc[0] + m15) * LDSLD + kk + kb];
          b0.y = Cb[(tc[0] + m15) * LDSLD + kk + kb + 1];
          acc[0] = wmma_f32_k4(a, b0, acc[0]);
          b1.x = Cb[(tc[1] + m15) * LDSLD + kk + kb];
          b1.y = Cb[(tc[1] + m15) * LDSLD + kk + kb + 1];
          acc[1] = wmma_f32_k4(a, b1, acc[1]);   // tr[0] == tr[1]
        }
        __syncthreads();
      }
      // ---- block-diagonal term: relu(h tile) @ Bm[j,j], K=64
      for (int i = tid; i < 64 * 64; i += 256) {
        const int r = i >> 6, c = i & 63;
        Sa[r * LDSLD + c] = Sprev[(size_t)r * H_DIM + jc + c];   // raw (epilogue)
        Cb[c * LDSLD + r] = Bm[(size_t)(jc + r) * H_DIM + jc + c];
      }
      __syncthreads();
      for (int kk = 0; kk < 64; kk += 4) {
        v2f a, b0, b1;
        a.x  = fmaxf(Sa[(tr[0] + m15) * LDSLD + kk + kb], 0.f);   // relu
        a.y  = fmaxf(Sa[(tr[0] + m15) * LDSLD + kk + kb + 1], 0.f);
        b0.x = Cb[(tc[0] + m15) * LDSLD + kk + kb];
        b0.y = Cb[(tc[0] + m15) * LDSLD + kk + kb + 1];
        acc[0] = wmma_f32_k4(a, b0, acc[0]);
        b1.x = Cb[(tc[1] + m15) * LDSLD + kk + kb];
        b1.y = Cb[(tc[1] + m15) * LDSLD + kk + kb + 1];
        acc[1] = wmma_f32_k4(a, b1, acc[1]);
      }
      // no sync needed: Sa stays read-only (raw h tile) until the epilogue
    }

    // ---- fused Euler update, in place over the proj slice we own
    float* Pt = emb + (size_t)t * BATCH * H_DIM;
#pragma unroll
    for (int s = 0; s < 2; ++s) {
#pragma unroll
      for (int r = 0; r < 8; ++r) {
        const int mm = tr[s] + r + mhi;
        const int nn = tc[s] + m15;
        const size_t g = (size_t)mm * H_DIM + jc + nn;
        const float p = Pt[g];                        // u_t element
        float o;
        if (t > 0) {
          const float sv = Sa[mm * LDSLD + nn];       // h_{t-1} element
          o = (1.f - EUL) * sv + EUL * (acc[s][r] + p);
        } else {
          o = EUL * p;                                // h_0 = 0
        }
        Pt[g] = o;                                    // becomes embeddings[t]
      }
    }

    // ---- device-scope barrier across the 16 resident workgroups
    __threadfence();
    __syncthreads();
    if (tid == 0) {
      __hip_atomic_fetch_add(&cnt[t], 1u, __ATOMIC_ACQ_REL,
                             __HIP_MEMORY_SCOPE_AGENT);
      while (__hip_atomic_load(&cnt[t], __ATOMIC_ACQUIRE,
                               __HIP_MEMORY_SCOPE_AGENT) < NWG) {
        __builtin_amdgcn_s_sleep(2);
      }
    }
    __syncthreads();
  }
}

__global__ void barrier_init(unsigned* __restrict__ cnt) {
  const int i = blockIdx.x * blockDim.x + threadIdx.x;
  if (i < T_STEPS) cnt[i] = 0u;
}

extern "C" void kernel_launch(void* const* d_in, const int* in_sizes, int n_in,
                              void* d_out, int out_size, void* d_ws,
                              size_t ws_size, hipStream_t stream) {
  const float* input     = (const float*)d_in[0];  // [T,B,IN]
  const float* input_mat = (const float*)d_in[1];  // [IN,H]
  const float* block_mat = (const float*)d_in[2];  // [H,H]
  const float* coupling  = (const float*)d_in[3];  // [H,H]
  const float* out_mat   = (const float*)d_in[4];  // [H,OUT]

  float* out = (float*)d_out;                                   // [T,B,OUT]
  float* emb = out + (size_t)T_STEPS * BATCH * OUT_DIM;         // [T,B,H]
  unsigned* cnt = (unsigned*)d_ws;                              // T counters

  // reset per-step barrier counters (d_ws is not re-poisoned between replays)
  barrier_init<<<(T_STEPS + 255) / 256, 256, 0, stream>>>(cnt);

  // proj = input @ W_in, written into the embeddings region (overwritten by scan)
  {
    dim3 grid(T_STEPS * BATCH / 64, H_DIM / 64);
    gemm64_f32<<<grid, 256, 0, stream>>>(input, input_mat, emb,
                                         T_STEPS * BATCH, H_DIM, IN_DIM,
                                         IN_DIM, H_DIM, H_DIM);
  }

  // sequential Euler scan: 16 persistent workgroups, one H-slice each
  rnn_scan<<<NWG, 256, 0, stream>>>(block_mat, coupling, emb, cnt);

  // output = embeddings @ W_out
  {
    dim3 grid(T_STEPS * BATCH / 64, OUT_DIM / 64);
    gemm64_f32<<<grid, 256, 0, stream>>>(emb, out_mat, out,
                                         T_STEPS * BATCH, OUT_DIM, H_DIM,
                                         H_DIM, OUT_DIM, OUT_DIM);
  }
}